// EventDrivenSMLP_37005438222902
// MI455X (gfx1250) — compile-verified
//
#include <hip/hip_runtime.h>
#include <stdint.h>

#define D_IN   784
#define D_H    400
#define D_OUT  10
#define NB     256
#define NE     1024
#define NWAVES 13            // 416 threads / 32
#define LOG2_DECAY (-2.3219280948873623f)   // log2(0.2); exp(-dt/TAU) == 0.2^dt == exp2(dt*log2(0.2))
#define THRESH 0.5f
#define TW_INV (1.0f / 64.0f)

// ---------- prep: W1T[p][j] = W1[j][p]  (coalesced row gathers later) ----------
__global__ void snn_prep_w1t(const float* __restrict__ W1, float* __restrict__ W1T) {
    int idx = blockIdx.x * blockDim.x + threadIdx.x;     // idx = p*400 + j
    if (idx < D_IN * D_H) {
        int p = idx / D_H;
        int j = idx - p * D_H;
        W1T[idx] = W1[j * D_IN + p];
    }
}

// ---------- prep: W2c[j][k] = W2[k][j] + b2[k]/400 ----------
__global__ void snn_prep_w2c(const float* __restrict__ W2, const float* __restrict__ b2,
                             float* __restrict__ W2c) {
    int idx = blockIdx.x * blockDim.x + threadIdx.x;     // idx = j*10 + k
    if (idx < D_H * D_OUT) {
        int j = idx / D_OUT;
        int k = idx - j * D_OUT;
        W2c[idx] = W2[k * D_H + j] + b2[k] * (1.0f / 400.0f);
    }
}

// ---------- main: one block per sample; event-driven LIF scan ----------
__launch_bounds__(416, 1)
__global__ void snn_main(const float* __restrict__ times, const int* __restrict__ pixels,
                         const float* __restrict__ b1,
                         const float* __restrict__ W1T, const float* __restrict__ W2c,
                         float* __restrict__ out) {
    __shared__ float s_w2c[D_H * D_OUT];                       // 16000 B
    __shared__ __attribute__((aligned(16))) float s_wrow[2][D_H];  // double-buffered W1T row
    __shared__ float    s_times[NE];                           // 4 KB
    __shared__ int      s_pix[NE];                             // 4 KB
    __shared__ unsigned s_fired[16];                           // 13 used

    const int tid = threadIdx.x;
    const int b   = blockIdx.x;
    const float* bt = times  + (size_t)b * NE;
    const int*   bp = pixels + (size_t)b * NE;

    for (int i = tid; i < NE; i += 416) { s_times[i] = bt[i]; s_pix[i] = bp[i]; }
    for (int i = tid; i < D_H * D_OUT; i += 416) s_w2c[i] = W2c[i];

    float h1 = 0.0f, bias1 = 0.0f;
    if (tid < D_H) bias1 = b1[tid] * (1.0f / 784.0f);
    float h2 = 0.0f, cnt = 0.0f;
    float tprev = 0.0f;

    __syncthreads();   // s_pix ready before first prefetch

    // Prefetch first W1T row into buffer 0: 400 f32 = 100 x b128, threads 0..99.
    if (tid < 100) {
        int p0 = s_pix[0];
        unsigned laddr = (unsigned)(uintptr_t)(const void*)&s_wrow[0][tid * 4];
        unsigned long long g =
            (unsigned long long)(uintptr_t)(W1T + (size_t)p0 * D_H + tid * 4);
        asm volatile("global_load_async_to_lds_b128 %0, %1, off"
                     :: "v"(laddr), "v"(g) : "memory");
    }

    int buf = 0;
    for (int e = 0; e < NE; ++e) {
        asm volatile("s_wait_asynccnt 0" ::: "memory");  // issuers retire their prefetch
        __syncthreads();                                 // s_wrow[buf] now valid for all waves

        float t  = s_times[e];
        float d1 = __builtin_amdgcn_exp2f((t - tprev) * LOG2_DECAY);  // 0.2^(t - tprev)
        tprev = t;

        // ---- layer 1: decay + integrate + fire/reset (neuron per lane) ----
        bool fired = false;
        if (tid < D_H) {
            h1 = h1 * d1 + s_wrow[buf][tid] + bias1;
            fired = (h1 >= THRESH);
            if (fired) h1 = 0.0f;
        }
        unsigned ball = __builtin_amdgcn_ballot_w32(fired);
        if ((tid & 31) == 0) s_fired[tid >> 5] = ball;

        // ---- async prefetch next event's row into the other buffer ----
        {
            int pn = s_pix[(e + 1 < NE) ? (e + 1) : 0];
            if (tid < 100) {
                unsigned laddr = (unsigned)(uintptr_t)(const void*)&s_wrow[buf ^ 1][tid * 4];
                unsigned long long g =
                    (unsigned long long)(uintptr_t)(W1T + (size_t)pn * D_H + tid * 4);
                asm volatile("global_load_async_to_lds_b128 %0, %1, off"
                             :: "v"(laddr), "v"(g) : "memory");
            }
        }

        __syncthreads();   // fired masks visible

        // ---- layer 2: sequential in-order hidden-event scan, 10-wide across lanes ----
        // (th_e - th_{e-1} == t_e - t_{e-1} for e>=1; for e==0 h2 is 0 so d2 is irrelevant)
        if (tid < D_OUT) {
            h2 *= d1;
            #pragma unroll 1
            for (int wv = 0; wv < NWAVES; ++wv) {
                unsigned m = s_fired[wv];        // uniform across the 10 lanes
                while (m) {
                    int j = __builtin_ctz(m);
                    m &= m - 1;
                    h2 += s_w2c[(wv * 32 + j) * D_OUT + tid];
                    if (h2 >= THRESH) { cnt += 1.0f; h2 = 0.0f; }
                }
            }
        }
        buf ^= 1;
    }

    asm volatile("s_wait_asynccnt 0" ::: "memory");  // retire final (dummy) prefetch

    if (tid < D_OUT) out[b * D_OUT + tid] = cnt * TW_INV;
}

extern "C" void kernel_launch(void* const* d_in, const int* in_sizes, int n_in,
                              void* d_out, int out_size, void* d_ws, size_t ws_size,
                              hipStream_t stream) {
    (void)in_sizes; (void)n_in; (void)out_size; (void)ws_size;
    const float* times = (const float*)d_in[0];   // [256,1024]
    const int*   pix   = (const int*)  d_in[1];   // [256,1024]
    const float* W1    = (const float*)d_in[2];   // [400,784]
    const float* b1    = (const float*)d_in[3];   // [400]
    const float* W2    = (const float*)d_in[4];   // [10,400]
    const float* b2    = (const float*)d_in[5];   // [10]
    float* out = (float*)d_out;                   // [256,10]

    float* W1T = (float*)d_ws;                                          // 784*400 f32
    float* W2c = (float*)((char*)d_ws + (size_t)D_IN * D_H * sizeof(float)); // 400*10 f32

    snn_prep_w1t<<<(D_IN * D_H + 255) / 256, 256, 0, stream>>>(W1, W1T);
    snn_prep_w2c<<<(D_H * D_OUT + 255) / 256, 256, 0, stream>>>(W2, b2, W2c);
    snn_main<<<NB, 416, 0, stream>>>(times, pix, b1, W1T, W2c, out);
}